// GATModel_48034914238528
// MI455X (gfx1250) — compile-verified
//
#include <hip/hip_runtime.h>

typedef __attribute__((ext_vector_type(2))) float v2f;
typedef __attribute__((ext_vector_type(8))) float v8f;

#define NN 50000
#define NE 800000
#define NG 128
#define SLOPE 0.2f

// ---------------------------------------------------------------- utilities
__global__ void k_fill(float* __restrict__ p, float v, int n) {
  int i = blockIdx.x * blockDim.x + threadIdx.x;
  if (i < n) p[i] = v;
}

// pad feature [NN,10] -> [NN,12] (zero cols 10,11)
__global__ void k_pad_feat(const float* __restrict__ s, float* __restrict__ d) {
  int i = blockIdx.x * blockDim.x + threadIdx.x;   // n*12 + k
  if (i >= NN * 12) return;
  int n = i / 12, k = i - n * 12;
  d[i] = (k < 10) ? s[n * 10 + k] : 0.f;
}

// pad W1 [10,96] -> [12,96] (zero rows 10,11)
__global__ void k_pad_w1(const float* __restrict__ s, float* __restrict__ d) {
  int i = blockIdx.x * blockDim.x + threadIdx.x;   // k*96 + c
  if (i >= 12 * 96) return;
  d[i] = (i < 10 * 96) ? s[i] : 0.f;
}

// ------------------------------------------------------------- WMMA GEMM
// C[M,N] = act(A)[M,K] @ B[K,N].  M multiple of 16, K multiple of 4, N mult of 16.
// One wave computes one 16x16 C tile via 16x16x4 f32 WMMA; 4 waves/block.
template <int K, int N, bool RELU>
__global__ void k_gemm_wmma(const float* __restrict__ A, const float* __restrict__ B,
                            float* __restrict__ C) {
  const int lane = threadIdx.x & 31;
  const int wave = threadIdx.x >> 5;
  const int row0 = blockIdx.x * 16;
  const int col0 = (blockIdx.y * (blockDim.x >> 5) + wave) * 16;
  if (col0 >= N) return;            // wave-uniform: EXEC stays all-ones for WMMA
  const int half = lane >> 4;       // A: lanes 0-15 carry K=kk,kk+1; 16-31 carry kk+2,kk+3
  const int lr   = lane & 15;
  const float* Ap = A + (size_t)(row0 + lr) * K + 2 * half;   // 8B aligned
  const float* Bp = B + (size_t)(2 * half) * N + col0 + lr;
  v8f acc = {};
#pragma unroll
  for (int kk = 0; kk < K; kk += 4) {
    float2 av = *(const float2*)(Ap + kk);                    // global_load_b64
    v2f a, b;
    a.x = RELU ? fmaxf(av.x, 0.f) : av.x;
    a.y = RELU ? fmaxf(av.y, 0.f) : av.y;
    b.x = Bp[(size_t)kk * N];
    b.y = Bp[(size_t)(kk + 1) * N];
    acc = __builtin_amdgcn_wmma_f32_16x16x4_f32(false, a, false, b,
                                                (short)0, acc, false, false);
  }
#pragma unroll
  for (int v = 0; v < 8; ++v) {     // C layout: VGPR v -> row v (lanes 0-15), v+8 (16-31)
    C[(size_t)(row0 + v + 8 * half) * N + col0 + lr] = acc[v];
  }
}

// ------------------------------------------------- attention logits per node
__global__ void k_attn_logits(const float* __restrict__ z, const float* __restrict__ al,
                              const float* __restrict__ ar, float* __restrict__ el,
                              float* __restrict__ er, int H, int D) {
  int i = blockIdx.x * blockDim.x + threadIdx.x;   // n*H + h
  if (i >= NN * 3) return;
  int n = i / 3, h = i - n * 3;
  const float* zp  = z  + (size_t)n * H * D + (size_t)h * D;
  const float* alp = al + (size_t)h * D;
  const float* arp = ar + (size_t)h * D;
  float sl = 0.f, sr = 0.f;
  for (int d = 0; d < D; d += 4) {
    float4 zv = *(const float4*)(zp + d);
    sl += zv.x * alp[d] + zv.y * alp[d + 1] + zv.z * alp[d + 2] + zv.w * alp[d + 3];
    sr += zv.x * arp[d] + zv.y * arp[d + 1] + zv.z * arp[d + 2] + zv.w * arp[d + 3];
  }
  el[i] = sl; er[i] = sr;
}

// --------------------------------------- edge pass 1: logits + segment max
__global__ void k_edge_logit_max(const float* __restrict__ el, const float* __restrict__ er,
                                 const int* __restrict__ src, const int* __restrict__ dst,
                                 float* __restrict__ ebuf, float* __restrict__ mx) {
  int e = blockIdx.x * blockDim.x + threadIdx.x;
  if (e >= NE) return;
  int s = src[e], d = dst[e];
#pragma unroll
  for (int h = 0; h < 3; ++h) {
    float v = el[s * 3 + h] + er[d * 3 + h];
    v = v > 0.f ? v : SLOPE * v;                   // leaky relu
    ebuf[(size_t)e * 3 + h] = v;
    atomicMax(&mx[d * 3 + h], v);
  }
}

// --------------------------------------- edge pass 2: exp + segment sum
__global__ void k_edge_exp_sum(const int* __restrict__ dst, float* __restrict__ ebuf,
                               const float* __restrict__ mx, float* __restrict__ ssum) {
  int e = blockIdx.x * blockDim.x + threadIdx.x;
  if (e >= NE) return;
  int d = dst[e];
#pragma unroll
  for (int h = 0; h < 3; ++h) {
    float ex = __expf(ebuf[(size_t)e * 3 + h] - mx[d * 3 + h]);
    ebuf[(size_t)e * 3 + h] = ex;
    atomicAdd(&ssum[d * 3 + h], ex);
  }
}

// --------------------------------------- edge pass 3: weighted aggregation
template <int D>
__global__ void k_edge_aggregate(const float* __restrict__ z, const float* __restrict__ ebuf,
                                 const float* __restrict__ ssum, const int* __restrict__ src,
                                 const int* __restrict__ dst, float* __restrict__ out) {
  int i = blockIdx.x * blockDim.x + threadIdx.x;   // e*3 + h
  if (i >= NE * 3) return;
  int e = i / 3, h = i - e * 3;
  int s = src[e], d = dst[e];
  float a = ebuf[(size_t)e * 3 + h] / (ssum[d * 3 + h] + 1e-16f);
  const float* zp = z   + (size_t)s * 3 * D + (size_t)h * D;
  float*       op = out + (size_t)d * 3 * D + (size_t)h * D;
  __builtin_prefetch(zp, 0, 1);                    // global_prefetch (rows live in L2)
#pragma unroll
  for (int dd = 0; dd < D; dd += 4) {
    float4 zv = *(const float4*)(zp + dd);         // global_load_b128 gather
    atomicAdd(&op[dd + 0], zv.x * a);
    atomicAdd(&op[dd + 1], zv.y * a);
    atomicAdd(&op[dd + 2], zv.z * a);
    atomicAdd(&op[dd + 3], zv.w * a);
  }
}

// ------------------------------------------------------- graph mean pool
__global__ void k_pool(const float* __restrict__ x2, const int* __restrict__ gid,
                       float* __restrict__ sums, float* __restrict__ cnt) {
  int n = blockIdx.x * blockDim.x + threadIdx.x;
  if (n >= NN) return;
  int g = gid[n];
  atomicAdd(&cnt[g], 1.f);
  const float* xp = x2 + (size_t)n * 192;
  float* sp = sums + (size_t)g * 192;
#pragma unroll 4
  for (int c = 0; c < 192; c += 4) {
    float4 v = *(const float4*)(xp + c);
    atomicAdd(&sp[c + 0], fmaxf(v.x, 0.f));        // relu on layer-2 output
    atomicAdd(&sp[c + 1], fmaxf(v.y, 0.f));
    atomicAdd(&sp[c + 2], fmaxf(v.z, 0.f));
    atomicAdd(&sp[c + 3], fmaxf(v.w, 0.f));
  }
}

// ------------------------------------------------------- dense head (tiny)
__global__ void k_head(const float* __restrict__ sums, const float* __restrict__ cnt,
                       const float* __restrict__ w1, const float* __restrict__ b1,
                       const float* __restrict__ w2, const float* __restrict__ b2,
                       float* __restrict__ out) {
  int g = threadIdx.x;
  if (g >= NG) return;
  float c = cnt[g]; c = fmaxf(c, 1.f);
  float inv = 1.f / c;
  float acc = b2[0];
  for (int j = 0; j < 64; ++j) {
    float hs = b1[j];
    for (int k = 0; k < 192; ++k) hs += sums[(size_t)g * 192 + k] * inv * w1[(size_t)k * 64 + j];
    acc += fmaxf(hs, 0.f) * w2[j];
  }
  out[g] = acc;
}

// --------------------------------------------------------------- launcher
extern "C" void kernel_launch(void* const* d_in, const int* in_sizes, int n_in,
                              void* d_out, int out_size, void* d_ws, size_t ws_size,
                              hipStream_t stream) {
  const float* feature = (const float*)d_in[0];
  const int*   src     = (const int*)  d_in[1];
  const int*   dst     = (const int*)  d_in[2];
  const int*   gid     = (const int*)  d_in[3];
  const float* W1      = (const float*)d_in[4];
  const float* al1     = (const float*)d_in[5];
  const float* ar1     = (const float*)d_in[6];
  const float* W2      = (const float*)d_in[7];
  const float* al2     = (const float*)d_in[8];
  const float* ar2     = (const float*)d_in[9];
  const float* d1w     = (const float*)d_in[10];
  const float* d1b     = (const float*)d_in[11];
  const float* d2w     = (const float*)d_in[12];
  const float* d2b     = (const float*)d_in[13];
  float* out = (float*)d_out;

  float* ws = (float*)d_ws;
  size_t off = 0;
  float* zbuf  = ws + off; off += (size_t)NN * 192;  // z1 (first 96 cols) then z2
  float* obuf  = ws + off; off += (size_t)NN * 192;  // out1 (first 96 cols) then out2
  float* ebuf  = ws + off; off += (size_t)NE * 3;
  float* el    = ws + off; off += (size_t)NN * 3;
  float* er    = ws + off; off += (size_t)NN * 3;
  float* mx    = ws + off; off += (size_t)NN * 3;
  float* ssum  = ws + off; off += (size_t)NN * 3;
  float* sums  = ws + off; off += (size_t)NG * 192;
  float* cnt   = ws + off; off += (size_t)NG;
  float* featp = ws + off; off += (size_t)NN * 12;   // K-padded feature
  float* w1p   = ws + off; off += (size_t)12 * 96;   // K-padded W1

  const int TB = 256;
  #define FILL(p, v, n) k_fill<<<(int)(((n) + TB - 1) / TB), TB, 0, stream>>>((p), (v), (int)(n))

  // ======================= Layer 1: GAT(10 -> 3x32) =======================
  k_pad_feat<<<(NN * 12 + TB - 1) / TB, TB, 0, stream>>>(feature, featp);
  k_pad_w1<<<(12 * 96 + TB - 1) / TB, TB, 0, stream>>>(W1, w1p);
  {
    dim3 grid(NN / 16, 2);                           // 6 col tiles, 4 waves/block
    k_gemm_wmma<12, 96, false><<<grid, 128, 0, stream>>>(featp, w1p, zbuf);
  }
  k_attn_logits<<<(NN * 3 + TB - 1) / TB, TB, 0, stream>>>(zbuf, al1, ar1, el, er, 3, 32);
  FILL(mx, -1e30f, (size_t)NN * 3);
  FILL(ssum, 0.f,  (size_t)NN * 3);
  FILL(obuf, 0.f,  (size_t)NN * 96);
  k_edge_logit_max<<<(NE + TB - 1) / TB, TB, 0, stream>>>(el, er, src, dst, ebuf, mx);
  k_edge_exp_sum  <<<(NE + TB - 1) / TB, TB, 0, stream>>>(dst, ebuf, mx, ssum);
  k_edge_aggregate<32><<<(NE * 3 + TB - 1) / TB, TB, 0, stream>>>(zbuf, ebuf, ssum, src, dst, obuf);

  // ======================= Layer 2: GAT(96 -> 3x64) =======================
  {
    dim3 grid(NN / 16, 3);                           // 12 col tiles, 4 waves/block
    k_gemm_wmma<96, 192, true><<<grid, 128, 0, stream>>>(obuf, W2, zbuf);
  }
  k_attn_logits<<<(NN * 3 + TB - 1) / TB, TB, 0, stream>>>(zbuf, al2, ar2, el, er, 3, 64);
  FILL(mx, -1e30f, (size_t)NN * 3);
  FILL(ssum, 0.f,  (size_t)NN * 3);
  FILL(obuf, 0.f,  (size_t)NN * 192);                // out1 dead after GEMM2
  k_edge_logit_max<<<(NE + TB - 1) / TB, TB, 0, stream>>>(el, er, src, dst, ebuf, mx);
  k_edge_exp_sum  <<<(NE + TB - 1) / TB, TB, 0, stream>>>(dst, ebuf, mx, ssum);
  k_edge_aggregate<64><<<(NE * 3 + TB - 1) / TB, TB, 0, stream>>>(zbuf, ebuf, ssum, src, dst, obuf);

  // ======================= Pool + MLP head =======================
  FILL(sums, 0.f, (size_t)NG * 192);
  FILL(cnt,  0.f, (size_t)NG);
  k_pool<<<(NN + TB - 1) / TB, TB, 0, stream>>>(obuf, gid, sums, cnt);
  k_head<<<1, 128, 0, stream>>>(sums, cnt, d1w, d1b, d2w, d2b, out);
  #undef FILL
}